// DecoderRNN_68968584839833
// MI455X (gfx1250) — compile-verified
//
#include <hip/hip_runtime.h>
#include <math.h>

// Problem sizes (fixed by the reference)
#define Bn 64
#define Tn 64
#define Hn 1024
#define Vn 32000
#define Gn 3072   // 3*H

typedef float v2f __attribute__((ext_vector_type(2)));
typedef float v8f __attribute__((ext_vector_type(8)));
typedef int   v4i __attribute__((ext_vector_type(4)));

#ifndef __has_builtin
#define __has_builtin(x) 0
#endif

#if __has_builtin(__builtin_amdgcn_global_load_async_to_lds_b128)
#define ASYNC_LDS 1
#else
#define ASYNC_LDS 0
#endif

// 16B global -> LDS copy. Async (ASYNCcnt-tracked, no VGPR round trip) when the
// gfx1250 builtin is available, else a plain float4 bounce.
// Builtin signature (from compiler diagnostic): param0 = AS(1) int4*, non-const.
__device__ __forceinline__ void cp16_to_lds(float* dst_lds, const float* src_gbl) {
#if ASYNC_LDS
    __builtin_amdgcn_global_load_async_to_lds_b128(
        (__attribute__((address_space(1))) v4i*)(uintptr_t)src_gbl,
        (__attribute__((address_space(3))) v4i*)(uintptr_t)(unsigned)(uintptr_t)dst_lds,
        0, 0);
#else
    *reinterpret_cast<float4*>(dst_lds) = *reinterpret_cast<const float4*>(src_gbl);
#endif
}

// Wait until at most N async-LDS ops from this wave remain in flight.
__device__ __forceinline__ void cp_wait_upto4(void) {
#if ASYNC_LDS
#if __has_builtin(__builtin_amdgcn_s_wait_asynccnt)
    __builtin_amdgcn_s_wait_asynccnt(4);
#else
    asm volatile("s_wait_asynccnt 0x4" ::: "memory");
#endif
#endif
}
__device__ __forceinline__ void cp_wait_all(void) {
#if ASYNC_LDS
#if __has_builtin(__builtin_amdgcn_s_wait_asynccnt)
    __builtin_amdgcn_s_wait_asynccnt(0);
#else
    asm volatile("s_wait_asynccnt 0x0" ::: "memory");
#endif
#endif
}

// Exact-precision CDNA5 tensor op: D(16x16,f32) = A(16x4,f32) x B(4x16,f32) + C
__device__ __forceinline__ v8f wmma4(v2f a, v2f b, v8f c) {
    return __builtin_amdgcn_wmma_f32_16x16x4_f32(
        false, a, false, b, (short)0, c, false, false);
}

__device__ __forceinline__ float sigm(float x) { return 1.0f / (1.0f + __expf(-x)); }

// ---------------------------------------------------------------------------
// Kernel 1: GI[b*T+t, :] = relu(emb[tok(b,t)]) @ w_ih^T + b_ih
// 64x64 tile per WG, K staged in double-buffered LDS via async-to-LDS.
// Both operands row-major in LDS (pad 34): A[m][k], B[n][k] -> b64 DS fusions.
// ReLU applied at A-fragment read (lets the gather be a pure byte copy).
// ---------------------------------------------------------------------------
__global__ __launch_bounds__(256)
void gi_kernel(const int* __restrict__ target, const float* __restrict__ emb,
               const float* __restrict__ w_ih, const float* __restrict__ b_ih,
               float* __restrict__ GI) {
    __shared__ float smem[2 * (64 * 34 + 64 * 34)];   // double buffer, 34.8 KB

    const int tid  = threadIdx.x;
    const int lane = tid & 31;
    const int w    = tid >> 5;
    const int r0 = blockIdx.y * 64;
    const int c0 = blockIdx.x * 64;
    const int mt  = w & 3;
    const int nt0 = (w >> 2) * 2;
    const int mrow  = lane & 15;
    const int khalf = (lane >> 4) * 2;

    // Per-thread staging assignment (fixed across K-chunks): two rows each.
    const int srow = tid >> 3;        // 0..31
    const int sk4  = tid & 7;         // float4 index within 32-wide K chunk
    const int bb   = r0 >> 6;         // whole tile is one batch row (T=64)
    const int tokA = (srow == 0) ? 0 : target[bb * Tn + srow - 1];
    const int tokB = target[bb * Tn + srow + 31];           // row srow+32
    const float* gA0 = emb + (size_t)tokA * Hn + sk4 * 4;
    const float* gA1 = emb + (size_t)tokB * Hn + sk4 * 4;
    const float* gB0 = w_ih + (size_t)(c0 + srow) * Hn + sk4 * 4;
    const float* gB1 = w_ih + (size_t)(c0 + srow + 32) * Hn + sk4 * 4;

    v8f acc0 = {}; v8f acc1 = {};

    const int NC = Hn / 32;           // 32 K-chunks
    {
        float* As = smem; float* Bs = smem + 64 * 34;
        cp16_to_lds(&As[srow * 34 + sk4 * 4],        gA0);
        cp16_to_lds(&As[(srow + 32) * 34 + sk4 * 4], gA1);
        cp16_to_lds(&Bs[srow * 34 + sk4 * 4],        gB0);
        cp16_to_lds(&Bs[(srow + 32) * 34 + sk4 * 4], gB1);
    }
    for (int ch = 0; ch < NC; ++ch) {
        float* buf = smem + (ch & 1) * (2 * 64 * 34);
        float* As = buf; float* Bs = buf + 64 * 34;
        if (ch + 1 < NC) {
            const int kb = (ch + 1) * 32;
            float* nb = smem + ((ch + 1) & 1) * (2 * 64 * 34);
            float* nA = nb; float* nB = nb + 64 * 34;
            cp16_to_lds(&nA[srow * 34 + sk4 * 4],        gA0 + kb);
            cp16_to_lds(&nA[(srow + 32) * 34 + sk4 * 4], gA1 + kb);
            cp16_to_lds(&nB[srow * 34 + sk4 * 4],        gB0 + kb);
            cp16_to_lds(&nB[(srow + 32) * 34 + sk4 * 4], gB1 + kb);
            cp_wait_upto4();       // drain older chunk, keep newest in flight
        } else {
            cp_wait_all();
        }
        __syncthreads();
#pragma unroll
        for (int k4 = 0; k4 < 8; ++k4) {
            const int kk = k4 * 4 + khalf;
            v2f a;  a.x  = fmaxf(As[(mt * 16 + mrow) * 34 + kk], 0.f);
                    a.y  = fmaxf(As[(mt * 16 + mrow) * 34 + kk + 1], 0.f);
            v2f b0; b0.x = Bs[(nt0 * 16 + mrow) * 34 + kk];
                    b0.y = Bs[(nt0 * 16 + mrow) * 34 + kk + 1];
            v2f b1; b1.x = Bs[((nt0 + 1) * 16 + mrow) * 34 + kk];
                    b1.y = Bs[((nt0 + 1) * 16 + mrow) * 34 + kk + 1];
            acc0 = wmma4(a, b0, acc0);
            acc1 = wmma4(a, b1, acc1);
        }
        __syncthreads();           // done reading buf before it is re-staged
    }
#pragma unroll
    for (int j = 0; j < 8; ++j) {
        int m  = j + ((lane >> 4) << 3);
        int gr = r0 + mt * 16 + m;
        int gc0 = c0 + nt0 * 16 + (lane & 15);
        int gc1 = gc0 + 16;
        GI[(size_t)gr * Gn + gc0] = acc0[j] + b_ih[gc0];
        GI[(size_t)gr * Gn + gc1] = acc1[j] + b_ih[gc1];
    }
}

// ---------------------------------------------------------------------------
// Kernel 2 (x64 sequential launches): one GRU step.
// 16 WGs x 64 H-columns; local GEMM M=64, N=192 (r/z/n), K=1024 with K-chunks
// of 16 so the double-buffered stage fits under the gh exchange buffer.
// ---------------------------------------------------------------------------
__global__ __launch_bounds__(256)
void gru_step_kernel(int t, const float* __restrict__ enc_hidden,
                     const float* __restrict__ w_hh, const float* __restrict__ b_hh,
                     const float* __restrict__ GI, float* __restrict__ hseq,
                     float* __restrict__ hfinal_out) {
    // stage buffers: 2 * (A 64*18 + B 192*18) = 9216 floats, aliased under
    // gh exchange buffer 64*196 = 12544 floats (50 KB total).
    __shared__ float smem[64 * 196];
    float* gh = smem;

    const int tid  = threadIdx.x;
    const int lane = tid & 31;
    const int w    = tid >> 5;
    const int c0   = blockIdx.x * 64;
    const int mt   = w & 3;
    const int ntb  = (w >> 2) * 6;
    const int mrow  = lane & 15;
    const int khalf = (lane >> 4) * 2;

    // Staging assignment (fixed across chunks): chunk = 16 K.
    const int srow = tid >> 2;        // 0..63
    const int sk4  = tid & 3;         // float4 within 16-wide chunk
    const float* gA = ((t == 0) ? enc_hidden + (size_t)srow * Hn
                                : hseq + ((size_t)srow * Tn + (t - 1)) * Hn)
                      + sk4 * 4;
    const float* gB0 = w_hh + (size_t)(0 * Hn + c0 + srow) * Hn + sk4 * 4;
    const float* gB1 = w_hh + (size_t)(1 * Hn + c0 + srow) * Hn + sk4 * 4;
    const float* gB2 = w_hh + (size_t)(2 * Hn + c0 + srow) * Hn + sk4 * 4;

    v8f acc[6] = {};

    const int HALF = 64 * 18 + 192 * 18;   // 4608 floats per buffer
    const int NC = Hn / 16;                // 64 chunks
    {
        float* As = smem; float* Bs = smem + 64 * 18;
        cp16_to_lds(&As[srow * 18 + sk4 * 4], gA);
        cp16_to_lds(&Bs[srow * 18 + sk4 * 4],         gB0);
        cp16_to_lds(&Bs[(srow + 64) * 18 + sk4 * 4],  gB1);
        cp16_to_lds(&Bs[(srow + 128) * 18 + sk4 * 4], gB2);
    }
    for (int ch = 0; ch < NC; ++ch) {
        float* buf = smem + (ch & 1) * HALF;
        float* As = buf; float* Bs = buf + 64 * 18;
        if (ch + 1 < NC) {
            const int kb = (ch + 1) * 16;
            float* nb = smem + ((ch + 1) & 1) * HALF;
            float* nA = nb; float* nB = nb + 64 * 18;
            cp16_to_lds(&nA[srow * 18 + sk4 * 4], gA + kb);
            cp16_to_lds(&nB[srow * 18 + sk4 * 4],         gB0 + kb);
            cp16_to_lds(&nB[(srow + 64) * 18 + sk4 * 4],  gB1 + kb);
            cp16_to_lds(&nB[(srow + 128) * 18 + sk4 * 4], gB2 + kb);
            cp_wait_upto4();
        } else {
            cp_wait_all();
        }
        __syncthreads();
#pragma unroll
        for (int k4 = 0; k4 < 4; ++k4) {
            const int kk = k4 * 4 + khalf;
            v2f a; a.x = As[(mt * 16 + mrow) * 18 + kk];
                   a.y = As[(mt * 16 + mrow) * 18 + kk + 1];
#pragma unroll
            for (int j = 0; j < 6; ++j) {
                v2f b; b.x = Bs[((ntb + j) * 16 + mrow) * 18 + kk];
                       b.y = Bs[((ntb + j) * 16 + mrow) * 18 + kk + 1];
                acc[j] = wmma4(a, b, acc[j]);
            }
        }
        __syncthreads();
    }

    // acc -> gh LDS (stage buffers dead; region reused)
#pragma unroll
    for (int j = 0; j < 6; ++j)
#pragma unroll
        for (int e = 0; e < 8; ++e) {
            int m = e + ((lane >> 4) << 3);
            gh[(mt * 16 + m) * 196 + (ntb + j) * 16 + (lane & 15)] = acc[j][e];
        }
    __syncthreads();

    // Fused gate math for this WG's 64 columns
    for (int i = tid; i < 4096; i += 256) {
        int b = i >> 6, jl = i & 63;
        int cg = c0 + jl;
        size_t gib = ((size_t)b * Tn + t) * Gn;
        float gir = GI[gib + cg];
        float giz = GI[gib + Hn + cg];
        float gin = GI[gib + 2 * Hn + cg];
        float ghr = gh[b * 196 + jl]       + b_hh[cg];
        float ghz = gh[b * 196 + 64 + jl]  + b_hh[Hn + cg];
        float ghn = gh[b * 196 + 128 + jl] + b_hh[2 * Hn + cg];
        float r = sigm(gir + ghr);
        float z = sigm(giz + ghz);
        float n = tanhf(gin + r * ghn);
        float hp = (t == 0) ? enc_hidden[(size_t)b * Hn + cg]
                            : hseq[((size_t)b * Tn + t - 1) * Hn + cg];
        float hnew = (1.0f - z) * n + z * hp;
        hseq[((size_t)b * Tn + t) * Hn + cg] = hnew;
        if (t == Tn - 1) hfinal_out[(size_t)b * Hn + cg] = hnew;
    }
}

// ---------------------------------------------------------------------------
// Kernel 3: logits = h_seq @ out_w^T + out_b  (raw logits into d_out).
// Dominant GEMM [4096,1024]x[1024,32000]; same pipelined tiling as kernel 1.
// ---------------------------------------------------------------------------
__global__ __launch_bounds__(256)
void logits_kernel(const float* __restrict__ hseq, const float* __restrict__ out_w,
                   const float* __restrict__ out_b, float* __restrict__ out) {
    __shared__ float smem[2 * (64 * 34 + 64 * 34)];

    const int tid  = threadIdx.x;
    const int lane = tid & 31;
    const int w    = tid >> 5;
    const int r0 = blockIdx.y * 64;
    const int c0 = blockIdx.x * 64;
    const int mt  = w & 3;
    const int nt0 = (w >> 2) * 2;
    const int mrow  = lane & 15;
    const int khalf = (lane >> 4) * 2;

    const int srow = tid >> 3;
    const int sk4  = tid & 7;
    const float* gA0 = hseq  + (size_t)(r0 + srow) * Hn + sk4 * 4;
    const float* gA1 = hseq  + (size_t)(r0 + srow + 32) * Hn + sk4 * 4;
    const float* gB0 = out_w + (size_t)(c0 + srow) * Hn + sk4 * 4;
    const float* gB1 = out_w + (size_t)(c0 + srow + 32) * Hn + sk4 * 4;

    v8f acc0 = {}; v8f acc1 = {};

    const int NC = Hn / 32;
    {
        float* As = smem; float* Bs = smem + 64 * 34;
        cp16_to_lds(&As[srow * 34 + sk4 * 4],        gA0);
        cp16_to_lds(&As[(srow + 32) * 34 + sk4 * 4], gA1);
        cp16_to_lds(&Bs[srow * 34 + sk4 * 4],        gB0);
        cp16_to_lds(&Bs[(srow + 32) * 34 + sk4 * 4], gB1);
    }
    for (int ch = 0; ch < NC; ++ch) {
        float* buf = smem + (ch & 1) * (2 * 64 * 34);
        float* As = buf; float* Bs = buf + 64 * 34;
        if (ch + 1 < NC) {
            const int kb = (ch + 1) * 32;
            float* nb = smem + ((ch + 1) & 1) * (2 * 64 * 34);
            float* nA = nb; float* nB = nb + 64 * 34;
            cp16_to_lds(&nA[srow * 34 + sk4 * 4],        gA0 + kb);
            cp16_to_lds(&nA[(srow + 32) * 34 + sk4 * 4], gA1 + kb);
            cp16_to_lds(&nB[srow * 34 + sk4 * 4],        gB0 + kb);
            cp16_to_lds(&nB[(srow + 32) * 34 + sk4 * 4], gB1 + kb);
            cp_wait_upto4();
        } else {
            cp_wait_all();
        }
        __syncthreads();
#pragma unroll
        for (int k4 = 0; k4 < 8; ++k4) {
            const int kk = k4 * 4 + khalf;
            v2f a;  a.x  = As[(mt * 16 + mrow) * 34 + kk];
                    a.y  = As[(mt * 16 + mrow) * 34 + kk + 1];
            v2f b0; b0.x = Bs[(nt0 * 16 + mrow) * 34 + kk];
                    b0.y = Bs[(nt0 * 16 + mrow) * 34 + kk + 1];
            v2f b1; b1.x = Bs[((nt0 + 1) * 16 + mrow) * 34 + kk];
                    b1.y = Bs[((nt0 + 1) * 16 + mrow) * 34 + kk + 1];
            acc0 = wmma4(a, b0, acc0);
            acc1 = wmma4(a, b1, acc1);
        }
        __syncthreads();
    }
#pragma unroll
    for (int j = 0; j < 8; ++j) {
        int m  = j + ((lane >> 4) << 3);
        int gr = r0 + mt * 16 + m;
        int gc0 = c0 + nt0 * 16 + (lane & 15);
        int gc1 = gc0 + 16;
        out[(size_t)gr * Vn + gc0] = acc0[j] + out_b[gc0];
        out[(size_t)gr * Vn + gc1] = acc1[j] + out_b[gc1];
    }
}

// ---------------------------------------------------------------------------
// Kernel 4: in-place log_softmax over V per (b,t) row (rows stay L2-hot).
// ---------------------------------------------------------------------------
__global__ __launch_bounds__(256)
void logsoftmax_kernel(float* __restrict__ out) {
    __shared__ float red[256];
    const int tid = threadIdx.x;
    float* p = out + (size_t)blockIdx.x * Vn;

    float mx = -INFINITY;
    for (int i = tid; i < Vn; i += 256) mx = fmaxf(mx, p[i]);
    red[tid] = mx; __syncthreads();
    for (int s = 128; s > 0; s >>= 1) {
        if (tid < s) red[tid] = fmaxf(red[tid], red[tid + s]);
        __syncthreads();
    }
    mx = red[0]; __syncthreads();

    float sum = 0.0f;
    for (int i = tid; i < Vn; i += 256) sum += __expf(p[i] - mx);
    red[tid] = sum; __syncthreads();
    for (int s = 128; s > 0; s >>= 1) {
        if (tid < s) red[tid] += red[tid + s];
        __syncthreads();
    }
    float lse = mx + logf(red[0]);
    __syncthreads();

    for (int i = tid; i < Vn; i += 256) p[i] -= lse;
}

// ---------------------------------------------------------------------------
extern "C" void kernel_launch(void* const* d_in, const int* in_sizes, int n_in,
                              void* d_out, int out_size, void* d_ws, size_t ws_size,
                              hipStream_t stream) {
    (void)in_sizes; (void)n_in; (void)out_size; (void)ws_size;
    const float* enc_hidden = (const float*)d_in[1];
    const int*   target     = (const int*)d_in[2];
    const float* emb   = (const float*)d_in[3];
    const float* w_ih  = (const float*)d_in[4];
    const float* w_hh  = (const float*)d_in[5];
    const float* b_ih  = (const float*)d_in[6];
    const float* b_hh  = (const float*)d_in[7];
    const float* out_w = (const float*)d_in[8];
    const float* out_b = (const float*)d_in[9];

    float* out    = (float*)d_out;
    float* GI     = (float*)d_ws;                 // [B*T, 3H] = 48 MiB
    float* hseq   = GI + (size_t)Bn * Tn * Gn;    // [B, T, H] = 16 MiB
    float* hfinal = out + (size_t)Bn * Tn * Vn;   // tail of d_out

    gi_kernel<<<dim3(Gn / 64, (Bn * Tn) / 64), 256, 0, stream>>>(
        target, emb, w_ih, b_ih, GI);

    for (int t = 0; t < Tn; ++t)
        gru_step_kernel<<<Hn / 64, 256, 0, stream>>>(
            t, enc_hidden, w_hh, b_hh, GI, hseq, hfinal);

    logits_kernel<<<dim3(Vn / 64, (Bn * Tn) / 64), 256, 0, stream>>>(
        hseq, out_w, out_b, out);

    logsoftmax_kernel<<<Bn * Tn, 256, 0, stream>>>(out);
}